// OrthogonalEmbeddings_27857157881981
// MI455X (gfx1250) — compile-verified
//
#include <hip/hip_runtime.h>
#include <stdint.h>

#define NUM_CATEGORIES 1024
#define TOKEN_DIM      1024
#define ROWS_PER_BLOCK 8   // 8 wave32's per 256-thread block, one row per wave

typedef __attribute__((ext_vector_type(4))) float f32x4;

__global__ __launch_bounds__(256) void onehot_gather_kernel(
    const float* __restrict__ x,      // [batch, NUM_CATEGORIES] one-hot f32
    const float* __restrict__ basis,  // [NUM_CATEGORIES, TOKEN_DIM] f32
    float* __restrict__ out,          // [batch, TOKEN_DIM] f32
    int batch)
{
    // 8 rows * 4 KB = 32 KB LDS per block (well under 320 KB/WGP)
    __shared__ __align__(16) float lds_rows[ROWS_PER_BLOCK][NUM_CATEGORIES];

    const int lane = threadIdx.x & 31;   // wave32
    const int wave = threadIdx.x >> 5;
    int row = blockIdx.x * ROWS_PER_BLOCK + wave;
    if (row >= batch) return;
    // row is uniform within the wave -> pin it to an SGPR so all row-base
    // address math below is scalar (saddr-form VMEM).
    row = __builtin_amdgcn_readfirstlane(row);

    // ---------- Phase 1: async-copy this wave's one-hot row into LDS ----------
    // 8 x global_load_async_to_lds_b128: 32 lanes * 16B = 512B per op -> 4KB row.
    // INST_OFFSET is added to BOTH the LDS destination and the global source
    // (CDNA5 ISA), so all 8 ops share one lds-offset VGPR and one
    // 64-bit address VGPR pair and differ only in the immediate offset.
    // th:TH_LOAD_NT: the 256MB one-hot stream must not evict the 4MB basis
    // from the 192MB L2.
    {
        const unsigned lds_off = (unsigned)(uintptr_t)&lds_rows[wave][0]
                               + (unsigned)lane * 16u;
        const uint64_t gaddr   = (uint64_t)(uintptr_t)
                                 (x + (size_t)row * NUM_CATEGORIES + lane * 4);
#define OHG_ASYNC_B128(OFF)                                                     \
        asm volatile("global_load_async_to_lds_b128 %0, %1, off offset:" #OFF   \
                     " th:TH_LOAD_NT"                                           \
                     :: "v"(lds_off), "v"(gaddr) : "memory")
        OHG_ASYNC_B128(0);    OHG_ASYNC_B128(512);
        OHG_ASYNC_B128(1024); OHG_ASYNC_B128(1536);
        OHG_ASYNC_B128(2048); OHG_ASYNC_B128(2560);
        OHG_ASYNC_B128(3072); OHG_ASYNC_B128(3584);
#undef OHG_ASYNC_B128
        asm volatile("s_wait_asynccnt 0" ::: "memory");
    }

    // ---------- Phase 2: scan the row in LDS for the hot column ----------
    unsigned idx = 0;
    #pragma unroll
    for (int i = 0; i < 8; ++i) {
        const int col = i * 128 + lane * 4;
        f32x4 v = *(const f32x4*)&lds_rows[wave][col];
        if (v.x != 0.0f) idx = (unsigned)col;
        if (v.y != 0.0f) idx = (unsigned)col + 1u;
        if (v.z != 0.0f) idx = (unsigned)col + 2u;
        if (v.w != 0.0f) idx = (unsigned)col + 3u;
    }
    // Exactly one lane/iteration saw the 1.0 -> OR-reduce across the wave32.
    #pragma unroll
    for (int off = 16; off > 0; off >>= 1)
        idx |= (unsigned)__shfl_xor((int)idx, off, 32);
    // Wave-uniform -> SGPR, so the basis row base is scalar.
    idx = (unsigned)__builtin_amdgcn_readfirstlane((int)idx);

    // ---------- Phase 3: gather basis[idx, :] -> out[row, :] ----------
    // basis reads hit L2 (4MB resident); output stores are non-temporal stream.
    const f32x4* __restrict__ brow = (const f32x4*)(basis + (size_t)idx * TOKEN_DIM);
    f32x4* __restrict__ orow       = (f32x4*)(out + (size_t)row * TOKEN_DIM);
    #pragma unroll
    for (int i = 0; i < 8; ++i) {
        f32x4 v = brow[i * 32 + lane];
        __builtin_nontemporal_store(v, &orow[i * 32 + lane]);
    }
}

extern "C" void kernel_launch(void* const* d_in, const int* in_sizes, int n_in,
                              void* d_out, int out_size, void* d_ws, size_t ws_size,
                              hipStream_t stream) {
    const float* x     = (const float*)d_in[0];  // [BATCH, NUM_CATEGORIES] f32
    const float* basis = (const float*)d_in[1];  // [NUM_CATEGORIES, TOKEN_DIM] f32
    float* out         = (float*)d_out;          // [BATCH, TOKEN_DIM] f32

    const int batch = in_sizes[0] / NUM_CATEGORIES;
    const int grid  = (batch + ROWS_PER_BLOCK - 1) / ROWS_PER_BLOCK;

    onehot_gather_kernel<<<dim3(grid), dim3(256), 0, stream>>>(x, basis, out, batch);
}